// CTSFNOBlock_30751965839470
// MI455X (gfx1250) — compile-verified
//
#include <hip/hip_runtime.h>
#include <hip/hip_bf16.h>
#include <math.h>

#define PI_D 3.14159265358979323846

// Padded dims (multiples of 16/32 so every GEMM is edge-free):
//   phi_in 255->256 (PP), m-bins 127->128 (MB), ext-theta 255->256 (TE),
//   phi_out 127->128 (PO). Zero pad rows/cols in generated tables make the
//   pad lanes contribute exactly 0.
#define PP 256
#define MB 128
#define TE 256
#define PO 128

typedef float v2f __attribute__((ext_vector_type(2)));
typedef float v8f __attribute__((ext_vector_type(8)));

// ---------------------------------------------------------------------------
// fp32 WMMA 16x16x4: D = A(16x4) * B(4x16) + C
// ---------------------------------------------------------------------------
__device__ __forceinline__ v8f wmma4(v2f a, v2f b, v8f c) {
  return __builtin_amdgcn_wmma_f32_16x16x4_f32(
      false, a, false, b, (short)0, c, false, false);
}

template <int MODE>
__device__ __forceinline__ void mma_step(v2f a1, v2f a2, v2f b1a, v2f b1b,
                                         v2f b2a, v2f b2b,
                                         v8f& acc1a, v8f& acc1b,
                                         v8f& acc2a, v8f& acc2b) {
  if (MODE == 0) {
    acc1a = wmma4(a1, b1a, acc1a);
    acc1b = wmma4(a1, b1b, acc1b);
  } else if (MODE == 1) {
    acc1a = wmma4(a1, b1a, acc1a);
    acc1b = wmma4(a1, b1b, acc1b);
    acc2a = wmma4(a2, b1a, acc2a);
    acc2b = wmma4(a2, b1b, acc2b);
  } else if (MODE == 2) {
    acc1a = wmma4(a1, b1a, acc1a);
    acc1b = wmma4(a1, b1b, acc1b);
    acc2a = wmma4(a1, b2a, acc2a);
    acc2b = wmma4(a1, b2b, acc2b);
  } else if (MODE == 3) {
    v2f a2n;
    a2n.x = -a2.x;
    a2n.y = -a2.y;
    acc1a = wmma4(a1, b1a, acc1a);   // +Ar*Br
    acc1a = wmma4(a2n, b2a, acc1a);  // -Ai*Bi
    acc1b = wmma4(a1, b1b, acc1b);
    acc1b = wmma4(a2n, b2b, acc1b);
    acc2a = wmma4(a1, b2a, acc2a);   // +Ar*Bi
    acc2a = wmma4(a2, b1a, acc2a);   // +Ai*Br
    acc2b = wmma4(a1, b2b, acc2b);
    acc2b = wmma4(a2, b1b, acc2b);
  } else {  // MODE 4
    acc1a = wmma4(a1, b1a, acc1a);
    acc1a = wmma4(a2, b2a, acc1a);
    acc1b = wmma4(a1, b1b, acc1b);
    acc1b = wmma4(a2, b2b, acc1b);
  }
}

// ---------------------------------------------------------------------------
// Generic strided-batched GEMM; one wave computes a 16x32 C tile (two 16x16
// WMMA tiles sharing the A fragment). Requires M%16==0, N%32==0, K%4==0.
// Tile bases are forced scalar (readfirstlane) so loads use the
// SADDR + 32-bit-voffset global form; per-lane offsets are small i32.
// MODE 0: C1 = A1@B1 (+bias)
// MODE 1: C1 = A1@B1 ; C2 = A2@B1            (dual-A, shared B)
// MODE 2: C1 = A1@B1 ; C2 = A1@B2            (shared-A, dual-B)
// MODE 3: C1 = A1@B1 - A2@B2 (+bias) ; C2 = A1@B2 + A2@B1   (complex)
// MODE 4: C1 = A1@B1 + A2@B2                 (dual-dot)
// A[b*sAb + m*sAm + k] ; B[b*sBb + k*sBk + n] ; C[b*sCb + m*sCm + n]
// ---------------------------------------------------------------------------
template <int MODE>
__global__ __launch_bounds__(128) void gemm_k(
    const float* __restrict__ A1, const float* __restrict__ A2,
    long long sAb, int sAm,
    const float* __restrict__ B1, const float* __restrict__ B2,
    long long sBb, int sBk,
    float* __restrict__ C1, float* __restrict__ C2,
    long long sCb, int sCm,
    const float* __restrict__ bias,
    int M, int N, int K, int nBatch) {
  constexpr bool HAS_A2 = (MODE == 1 || MODE == 3 || MODE == 4);
  constexpr bool HAS_B2 = (MODE == 2 || MODE == 3 || MODE == 4);
  constexpr bool HAS_C2 = (MODE == 1 || MODE == 2 || MODE == 3);

  const int lane = threadIdx.x & 31;
  const int wave = threadIdx.x >> 5;
  const int tilesM = M >> 4;
  const int tilesN = N >> 5;
  int gw = blockIdx.x * (blockDim.x >> 5) + wave;
  if (gw >= nBatch * tilesM * tilesN) return;  // wave-uniform
  gw = __builtin_amdgcn_readfirstlane(gw);     // force scalar tile indices

  const int b = gw / (tilesM * tilesN);
  const int tt = gw - b * (tilesM * tilesN);
  const int m0 = (tt / tilesN) << 4;
  const int n0 = (tt - (tt / tilesN) * tilesN) << 5;

  // Scalar (SGPR) tile bases
  const float* Ab1 = A1 + (long long)b * sAb + (long long)m0 * sAm;
  const float* Ab2 = HAS_A2 ? A2 + (long long)b * sAb + (long long)m0 * sAm : nullptr;
  const float* Bb1 = B1 + (long long)b * sBb + n0;
  const float* Bb2 = HAS_B2 ? B2 + (long long)b * sBb + n0 : nullptr;

  // Per-lane 32-bit element offsets.
  // A frag: lane&15 = M row, lane>>4 selects K pair {0,1}/{2,3}
  // B frag: lane&15 = N col, same K-pair split
  const int lrow = lane & 15;
  const int koff = (lane >> 4) << 1;
  int ao = lrow * sAm + koff;
  int bo = koff * sBk + lrow;
  const int bStep = 4 * sBk;

  v8f acc1a = {}, acc1b = {}, acc2a = {}, acc2b = {};

  // -- software pipeline: preload step 0 --
  v2f a1 = *(const v2f*)(Ab1 + ao);
  v2f b1a, b1b;
  b1a.x = Bb1[bo];
  b1a.y = Bb1[bo + sBk];
  b1b.x = Bb1[bo + 16];
  b1b.y = Bb1[bo + sBk + 16];
  v2f a2 = {}, b2a = {}, b2b = {};
  if (HAS_A2) a2 = *(const v2f*)(Ab2 + ao);
  if (HAS_B2) {
    b2a.x = Bb2[bo];
    b2a.y = Bb2[bo + sBk];
    b2b.x = Bb2[bo + 16];
    b2b.y = Bb2[bo + sBk + 16];
  }
  ao += 4;
  bo += bStep;

#pragma unroll 2
  for (int k0 = 4; k0 < K; k0 += 4) {
    // prefetch next step's fragments before consuming current ones
    v2f a1n = *(const v2f*)(Ab1 + ao);
    v2f b1an, b1bn;
    b1an.x = Bb1[bo];
    b1an.y = Bb1[bo + sBk];
    b1bn.x = Bb1[bo + 16];
    b1bn.y = Bb1[bo + sBk + 16];
    v2f a2n = {}, b2an = {}, b2bn = {};
    if (HAS_A2) a2n = *(const v2f*)(Ab2 + ao);
    if (HAS_B2) {
      b2an.x = Bb2[bo];
      b2an.y = Bb2[bo + sBk];
      b2bn.x = Bb2[bo + 16];
      b2bn.y = Bb2[bo + sBk + 16];
    }
    ao += 4;
    bo += bStep;

    mma_step<MODE>(a1, a2, b1a, b1b, b2a, b2b, acc1a, acc1b, acc2a, acc2b);

    a1 = a1n; a2 = a2n;
    b1a = b1an; b1b = b1bn;
    b2a = b2an; b2b = b2bn;
  }
  mma_step<MODE>(a1, a2, b1a, b1b, b2a, b2b, acc1a, acc1b, acc2a, acc2b);

  float bias0 = 0.f, bias1 = 0.f;
  if ((MODE == 0 || MODE == 3) && bias) {
    bias0 = bias[n0 + lrow];
    bias1 = bias[n0 + lrow + 16];
  }
  // C/D layout: VGPR r -> (M = m0 + r + 8*(lane>>4), N = n0 + (lane&15))
  float* Cb1 = C1 + (long long)b * sCb + (long long)m0 * sCm + n0;
  float* Cb2 = HAS_C2 ? C2 + (long long)b * sCb + (long long)m0 * sCm + n0 : nullptr;
  int co = ((lane >> 4) << 3) * sCm + lrow;
#pragma unroll
  for (int r = 0; r < 8; ++r) {
    Cb1[co] = acc1a[r] + bias0;
    Cb1[co + 16] = acc1b[r] + bias1;
    if (HAS_C2) {
      Cb2[co] = acc2a[r];
      Cb2[co + 16] = acc2b[r];
    }
    co += sCm;
  }
}

// ---------------------------------------------------------------------------
// Stage x into padded layout [128][PP][64], zero row p=255.
// ---------------------------------------------------------------------------
__global__ void xpad_k(const float* __restrict__ x, float* __restrict__ xp) {
  int idx = blockIdx.x * blockDim.x + threadIdx.x;
  if (idx >= 128 * PP * 64) return;
  int c = idx & 63;
  int rest = idx >> 6;
  int p = rest & (PP - 1);
  int t = rest >> 8;
  xp[idx] = (p < 255) ? x[(size_t)t * 255 * 64 + (size_t)p * 64 + c] : 0.f;
}

// Time-embedding projections + combined biases.
__global__ void tvec_k(const float* __restrict__ t_emb,
                       const float* __restrict__ trw, const float* __restrict__ trb,
                       const float* __restrict__ tiw, const float* __restrict__ tib,
                       const float* __restrict__ stw, const float* __restrict__ stb,
                       const float* __restrict__ c2b,
                       float* tr, float* ti, float* wt, float* bt, float* b2c) {
  int o = blockIdx.x * blockDim.x + threadIdx.x;
  if (o >= 64) return;
  float s0 = 0.f, s1 = 0.f, s2 = 0.f, s3 = 0.f;
  for (int d = 0; d < 256; ++d) {
    float e = t_emb[d];
    s0 += e * trw[d * 64 + o];
    s1 += e * tiw[d * 64 + o];
    s2 += e * stw[d * 128 + o];
    s3 += e * stw[d * 128 + 64 + o];
  }
  tr[o] = s0 + trb[o];
  ti[o] = s1 + tib[o];
  wt[o] = s2 + stb[o];
  float btv = s3 + stb[64 + o];
  bt[o] = btv;
  b2c[o] = btv + c2b[o];
}

// WR[l,i,o] = conv_w[i,o] + tr[l]*wr - ti[l]*wi ; WI[l,i,o] = tr[l]*wi + ti[l]*wr
__global__ void specw_k(const float* __restrict__ convw,
                        const float* __restrict__ wr, const float* __restrict__ wi,
                        const float* __restrict__ tr, const float* __restrict__ ti,
                        float* __restrict__ WR, float* __restrict__ WI) {
  int idx = blockIdx.x * blockDim.x + threadIdx.x;
  if (idx >= 64 * 64 * 64) return;
  int l = idx >> 12;
  int io = idx & 4095;
  float a = tr[l], b = ti[l];
  float r = wr[idx], q = wi[idx];
  WR[idx] = convw[io] + a * r - b * q;
  WI[idx] = a * q + b * r;
}

// Wc = conv1_w @ (wt[o] .* spat_weight) ; bc = conv1_b @ (...)
__global__ void spatw_k(const float* __restrict__ c1w, const float* __restrict__ c1b,
                        const float* __restrict__ sw, const float* __restrict__ wt,
                        float* __restrict__ Wc, float* __restrict__ bc) {
  int idx = blockIdx.x * blockDim.x + threadIdx.x;
  if (idx >= 64 * 64) return;
  int i = idx >> 6, o = idx & 63;
  float s = 0.f;
  for (int k = 0; k < 64; ++k) s += c1w[i * 64 + k] * sw[k * 64 + o];
  Wc[idx] = s * wt[o];
  if (i == 0) {
    float sb = 0.f;
    for (int k = 0; k < 64; ++k) sb += c1b[k] * sw[k * 64 + o];
    bc[o] = sb * wt[o];
  }
}

// Forward DFT matrices [MB][PP], m = mi-63; zero pad row mi=127 / col p=255.
__global__ void dftf_k(float* __restrict__ CfT, float* __restrict__ SfT) {
  int idx = blockIdx.x * blockDim.x + threadIdx.x;
  if (idx >= MB * PP) return;
  int mi = idx >> 8, p = idx & (PP - 1);
  if (mi == 127 || p == 255) {
    CfT[idx] = 0.f;
    SfT[idx] = 0.f;
    return;
  }
  double ang = -2.0 * PI_D * (double)(mi - 63) * (double)p / 255.0;
  double sc = 2.0 * PI_D / 255.0;
  CfT[idx] = (float)(sc * cos(ang));
  SfT[idx] = (float)(sc * sin(ang));
}

// Inverse DFT (real part) [PO][MB]; zero pad row p=127 / col mi=127.
__global__ void dfti_k(float* __restrict__ Ci, float* __restrict__ Si) {
  int idx = blockIdx.x * blockDim.x + threadIdx.x;
  if (idx >= PO * MB) return;
  int p = idx >> 7, mi = idx & (MB - 1);
  if (p == 127 || mi == 127) {
    Ci[idx] = 0.f;
    Si[idx] = 0.f;
    return;
  }
  double ang = 2.0 * PI_D * (double)(mi - 63) * (double)p / 127.0;
  Ci[idx] = (float)cos(ang);
  Si[idx] = (float)(-sin(ang));
}

// MW quadrature weights for L=128 over the 255 extended thetas.
__global__ void qw_k(float* __restrict__ qw) {
  int t = blockIdx.x * blockDim.x + threadIdx.x;
  if (t >= 255) return;
  double th = (2.0 * t + 1.0) * PI_D / 255.0;
  double s = 0.0;
  for (int k = -127; k <= 127; ++k) {
    double wr = 0.0, wi = 0.0;
    if (k == 1) wi = 0.5 * PI_D;
    else if (k == -1) wi = -0.5 * PI_D;
    else if ((k & 1) == 0) wr = 2.0 / (1.0 - (double)k * (double)k);
    s += wr * cos(th * k) - wi * sin(th * k);
  }
  qw[t] = (float)(s / 255.0);
}

// Seed d^{l0}_{m,-1}(theta), l0 = max(|m|,1), Varshalovich convention
// (matches the exp(-i*theta*Jy) eigen-construction of the reference).
__device__ __forceinline__ double wigner_seed(int m, double theta) {
  if (m == 0) return -sin(theta) * 0.70710678118654752440;
  int Mm = m > 0 ? m : -m;
  double c2 = cos(0.5 * theta), s2 = sin(0.5 * theta);
  double amp = exp(0.5 * (lgamma(2.0 * Mm + 1.0) - lgamma((double)Mm + 2.0) - lgamma((double)Mm)));
  if (m > 0) {
    double sign = ((Mm + 1) & 1) ? -1.0 : 1.0;  // (-sin)^{M+1}
    return sign * amp * pow(c2, (double)(Mm - 1)) * pow(s2, (double)(Mm + 1));
  }
  return amp * pow(c2, (double)(Mm + 1)) * pow(s2, (double)(Mm - 1));
}

// Three-term l-recurrence for d^l_{m,-1}. Fwd table [MB][64][TE] with MW
// quadrature weight and CL_IN folded in; zero pads at mi=127 / t=255.
__global__ void wigner_fwd_k(float* __restrict__ D, const float* __restrict__ qw) {
  int t = blockIdx.x * blockDim.x + threadIdx.x;
  int mi = blockIdx.y;
  if (t >= TE || mi >= MB) return;
  float* base = D + (size_t)mi * 64 * TE + t;
  if (mi == 127 || t == 255) {
    for (int l = 0; l < 64; ++l) base[(size_t)l * TE] = 0.f;
    return;
  }
  int m = mi - 63;
  double th = (2.0 * t + 1.0) * PI_D / 255.0;
  double c = cos(th);
  double w = (double)qw[t];
  int Mm = m > 0 ? m : -m;
  int l0 = Mm > 1 ? Mm : 1;
  double dp = 0.0, dc = wigner_seed(m, th);
  for (int l = 0; l < 64; ++l) {
    double val = (l < l0) ? 0.0 : dc;
    double cl = -sqrt((2.0 * l + 1.0) / (4.0 * PI_D));  // CL_IN (spin=1)
    base[(size_t)l * TE] = (float)(val * w * cl);
    if (l >= l0) {
      double L = (double)l;
      double num = (2.0 * L + 1.0) * (L * (L + 1.0) * c + (double)m) * dc -
                   (L + 1.0) * sqrt((L * L - (double)m * (double)m) * (L * L - 1.0)) * dp;
      double den = L * sqrt(((L + 1.0) * (L + 1.0) - (double)m * (double)m) *
                            ((L + 1.0) * (L + 1.0) - 1.0));
      double dn = num / den;
      dp = dc;
      dc = dn;
    }
  }
}

// Inverse table [MB][64][64] with CL_OUT folded in; zero pad at mi=127.
__global__ void wigner_inv_k(float* __restrict__ D) {
  int t = blockIdx.x * blockDim.x + threadIdx.x;
  int mi = blockIdx.y;
  if (t >= 64 || mi >= MB) return;
  float* base = D + (size_t)mi * 64 * 64 + (size_t)t * 64;
  if (mi == 127) {
    for (int l = 0; l < 64; ++l) base[l] = 0.f;
    return;
  }
  int m = mi - 63;
  double th = (2.0 * t + 1.0) * PI_D / 127.0;
  double c = cos(th);
  int Mm = m > 0 ? m : -m;
  int l0 = Mm > 1 ? Mm : 1;
  double dp = 0.0, dc = wigner_seed(m, th);
  for (int l = 0; l < 64; ++l) {
    double val = (l < l0) ? 0.0 : dc;
    double cl = -sqrt((2.0 * l + 1.0) / (4.0 * PI_D));  // CL_OUT (spin=1)
    base[l] = (float)(val * cl);
    if (l >= l0) {
      double L = (double)l;
      double num = (2.0 * L + 1.0) * (L * (L + 1.0) * c + (double)m) * dc -
                   (L + 1.0) * sqrt((L * L - (double)m * (double)m) * (L * L - 1.0)) * dp;
      double den = L * sqrt(((L + 1.0) * (L + 1.0) - (double)m * (double)m) *
                            ((L + 1.0) * (L + 1.0) - 1.0));
      double dn = num / den;
      dp = dc;
      dc = dn;
    }
  }
}

// g_ext[mi][t][c] (t<TE) from ftm[t'][mi][c] (t'<128) with parity extension.
// Zero pad at t>=255 or mi==127.
__global__ void gext_k(const float* __restrict__ fre, const float* __restrict__ fim,
                       float* __restrict__ gre, float* __restrict__ gim) {
  int idx = blockIdx.x * blockDim.x + threadIdx.x;
  if (idx >= MB * TE * 64) return;
  int c = idx & 63;
  int rest = idx >> 6;
  int t = rest & (TE - 1);
  int mi = rest >> 8;
  if (t >= 255 || mi == 127) {
    gre[idx] = 0.f;
    gim[idx] = 0.f;
    return;
  }
  int m = mi - 63;
  int ts;
  float sgn;
  if (t < 128) { ts = t; sgn = 1.f; }
  else { ts = 254 - t; sgn = (m & 1) ? 1.f : -1.f; }  // (-1)^(m+1), spin=1
  size_t src = (size_t)ts * MB * 64 + (size_t)mi * 64 + c;
  gre[idx] = sgn * fre[src];
  gim[idx] = sgn * fim[src];
}

// Per-row L2 normalization over 64 channels (one wave per row).
__global__ void rownorm_k(float* __restrict__ xs, int rows) {
  int row = blockIdx.x * (blockDim.x >> 5) + (threadIdx.x >> 5);
  int lane = threadIdx.x & 31;
  if (row >= rows) return;
  float* p = xs + (size_t)row * 64;
  float v0 = p[lane], v1 = p[lane + 32];
  float s = v0 * v0 + v1 * v1;
  for (int o = 16; o > 0; o >>= 1) s += __shfl_xor(s, o, 32);
  float inv = 1.f / (sqrtf(s) + 1e-6f);
  p[lane] = v0 * inv;
  p[lane + 32] = v1 * inv;
}

__device__ __forceinline__ float gelu_tanh(float x) {
  float x3 = x * x * x;
  return 0.5f * x * (1.f + tanhf(0.7978845608028654f * (x + 0.044715f * x3)));
}

// y = LayerNorm(gelu(xs + h)) * scale + bias; padded [64][PO][64] -> compact
// (64,127,64) output. One wave per output row.
__global__ void final_k(const float* __restrict__ xs, const float* __restrict__ h,
                        const float* __restrict__ lns, const float* __restrict__ lnb,
                        float* __restrict__ out) {
  int r = blockIdx.x * (blockDim.x >> 5) + (threadIdx.x >> 5);
  int lane = threadIdx.x & 31;
  if (r >= 64 * 127) return;
  int t = r / 127;
  int p = r - t * 127;
  size_t src = (size_t)t * PO * 64 + (size_t)p * 64;
  size_t dst = (size_t)r * 64;
  float y0 = gelu_tanh(xs[src + lane] + h[src + lane]);
  float y1 = gelu_tanh(xs[src + lane + 32] + h[src + lane + 32]);
  float s = y0 + y1, s2 = y0 * y0 + y1 * y1;
  for (int o = 16; o > 0; o >>= 1) {
    s += __shfl_xor(s, o, 32);
    s2 += __shfl_xor(s2, o, 32);
  }
  float mu = s * (1.f / 64.f);
  float var = s2 * (1.f / 64.f) - mu * mu;
  float inv = rsqrtf(var + 1e-6f);
  out[dst + lane] = (y0 - mu) * inv * lns[lane] + lnb[lane];
  out[dst + lane + 32] = (y1 - mu) * inv * lns[lane + 32] + lnb[lane + 32];
}

// ---------------------------------------------------------------------------
extern "C" void kernel_launch(void* const* d_in, const int* in_sizes, int n_in,
                              void* d_out, int out_size, void* d_ws, size_t ws_size,
                              hipStream_t stream) {
  const float* x       = (const float*)d_in[0];   // (128,255,64)
  const float* t_emb   = (const float*)d_in[1];   // (256,)
  const float* sconv_w = (const float*)d_in[2];   // (64,64)
  const float* sconv_b = (const float*)d_in[3];
  const float* swr     = (const float*)d_in[4];   // (64,64,64)
  const float* swi     = (const float*)d_in[5];
  const float* strw    = (const float*)d_in[6];   // (256,64)
  const float* strb    = (const float*)d_in[7];
  const float* stiw    = (const float*)d_in[8];
  const float* stib    = (const float*)d_in[9];
  const float* c1w     = (const float*)d_in[10];  // (64,64)
  const float* c1b     = (const float*)d_in[11];
  const float* sptw    = (const float*)d_in[12];  // (256,128)
  const float* sptb    = (const float*)d_in[13];  // (128,)
  const float* sweight = (const float*)d_in[14];  // (64,64)
  const float* c2w     = (const float*)d_in[15];  // (64,64)
  const float* c2b     = (const float*)d_in[16];
  const float* lns     = (const float*)d_in[17];
  const float* lnb     = (const float*)d_in[18];
  float* out           = (float*)d_out;           // (64,127,64)

  // Workspace bump allocator (256B aligned; ~74 MB total)
  char* wsb = (char*)d_ws;
  size_t off = 0;
  auto allocf = [&](size_t n) {
    float* p = (float*)(wsb + off);
    off += ((n * sizeof(float) + 255) / 256) * 256;
    return p;
  };
  float* xpad    = allocf((size_t)128 * PP * 64);
  float* CfT     = allocf((size_t)MB * PP);
  float* SfT     = allocf((size_t)MB * PP);
  float* Ci      = allocf((size_t)PO * MB);
  float* Si      = allocf((size_t)PO * MB);
  float* qw      = allocf(255);
  float* Dfwd    = allocf((size_t)MB * 64 * TE);
  float* Dinv    = allocf((size_t)MB * 64 * 64);
  float* tr      = allocf(64);
  float* ti      = allocf(64);
  float* wt      = allocf(64);
  float* bt      = allocf(64);
  float* b2c     = allocf(64);
  float* WR      = allocf((size_t)64 * 64 * 64);
  float* WI      = allocf((size_t)64 * 64 * 64);
  float* Wc      = allocf(64 * 64);
  float* bc      = allocf(64);
  float* ftm_re  = allocf((size_t)128 * MB * 64);
  float* ftm_im  = allocf((size_t)128 * MB * 64);
  float* g_re    = allocf((size_t)MB * TE * 64);
  float* g_im    = allocf((size_t)MB * TE * 64);
  float* flm_re  = allocf((size_t)64 * MB * 64);
  float* flm_im  = allocf((size_t)64 * MB * 64);
  float* flm2_re = allocf((size_t)64 * MB * 64);
  float* flm2_im = allocf((size_t)64 * MB * 64);
  float* ftm2_re = allocf((size_t)64 * MB * 64);
  float* ftm2_im = allocf((size_t)64 * MB * 64);
  float* xs      = allocf((size_t)64 * PO * 64);
  float* h1      = allocf((size_t)128 * PP * 64);
  float* hgrid   = allocf((size_t)64 * PO * 64);
  float* hfin    = allocf((size_t)64 * PO * 64);
  (void)ws_size; (void)in_sizes; (void)n_in; (void)out_size;

  auto gblocks = [](int M, int N, int nb) {
    int tiles = (M >> 4) * (N >> 5) * nb;
    return dim3((unsigned)((tiles + 3) / 4));
  };
  const dim3 GB(128);

  // ---- Precompute tables ----
  xpad_k<<<(128 * PP * 64) / 256, 256, 0, stream>>>(x, xpad);
  tvec_k<<<1, 64, 0, stream>>>(t_emb, strw, strb, stiw, stib, sptw, sptb, c2b,
                               tr, ti, wt, bt, b2c);
  specw_k<<<1024, 256, 0, stream>>>(sconv_w, swr, swi, tr, ti, WR, WI);
  spatw_k<<<16, 256, 0, stream>>>(c1w, c1b, sweight, wt, Wc, bc);
  dftf_k<<<(MB * PP) / 256, 256, 0, stream>>>(CfT, SfT);
  dfti_k<<<(PO * MB) / 256, 256, 0, stream>>>(Ci, Si);
  qw_k<<<1, 256, 0, stream>>>(qw);
  wigner_fwd_k<<<dim3(TE / 64, MB), 64, 0, stream>>>(Dfwd, qw);
  wigner_inv_k<<<dim3(1, MB), 64, 0, stream>>>(Dinv);

  // ---- Spectral path ----
  // ftm[t][mi][c] = {CfT,SfT} @ xpad[t]   (batch t=128, M=MB, K=PP, N=64)
  gemm_k<1><<<gblocks(MB, 64, 128), GB, 0, stream>>>(
      CfT, SfT, 0, PP, xpad, nullptr, PP * 64, 64,
      ftm_re, ftm_im, MB * 64, 64, nullptr, MB, 64, PP, 128);
  gext_k<<<(MB * TE * 64) / 256, 256, 0, stream>>>(ftm_re, ftm_im, g_re, g_im);
  // flm[l][mi][c] = Dfwd[mi] @ g[mi]      (batch mi=MB, M=64, K=TE, N=64)
  gemm_k<2><<<gblocks(64, 64, MB), GB, 0, stream>>>(
      Dfwd, nullptr, 64 * TE, TE, g_re, g_im, TE * 64, 64,
      flm_re, flm_im, 64, MB * 64, nullptr, 64, 64, TE, MB);
  // flm2[l][mi][o] = flm[l] (x) {WR,WI}[l] + conv_b  (batch l=64, M=MB, K=64, N=64)
  gemm_k<3><<<gblocks(MB, 64, 64), GB, 0, stream>>>(
      flm_re, flm_im, MB * 64, 64, WR, WI, 64 * 64, 64,
      flm2_re, flm2_im, MB * 64, 64, sconv_b, MB, 64, 64, 64);
  // ftm2[t][mi][o] = Dinv[mi] @ flm2[:,mi,:]  (batch mi=MB, M=64, K=64, N=64)
  gemm_k<2><<<gblocks(64, 64, MB), GB, 0, stream>>>(
      Dinv, nullptr, 64 * 64, 64, flm2_re, flm2_im, 64, MB * 64,
      ftm2_re, ftm2_im, 64, MB * 64, nullptr, 64, 64, 64, MB);
  // xs[t][p][o] = Ci @ ftm2_re[t] + Si @ ftm2_im[t]  (batch t=64, M=PO, K=MB, N=64)
  gemm_k<4><<<gblocks(PO, 64, 64), GB, 0, stream>>>(
      Ci, Si, 0, MB, ftm2_re, ftm2_im, MB * 64, 64,
      xs, nullptr, PO * 64, 64, nullptr, PO, 64, MB, 64);
  rownorm_k<<<(64 * PO) / 8, 256, 0, stream>>>(xs, 64 * PO);

  // ---- Spatial path ----
  // h1[t][p][o] = xpad[t] @ Wc + bc  (batch t=128, M=PP, K=64, N=64)
  gemm_k<0><<<gblocks(PP, 64, 128), GB, 0, stream>>>(
      xpad, nullptr, PP * 64, 64, Wc, nullptr, 0, 64,
      h1, nullptr, PP * 64, 64, bc, PP, 64, 64, 128);
  gemm_k<1><<<gblocks(MB, 64, 128), GB, 0, stream>>>(
      CfT, SfT, 0, PP, h1, nullptr, PP * 64, 64,
      ftm_re, ftm_im, MB * 64, 64, nullptr, MB, 64, PP, 128);
  gext_k<<<(MB * TE * 64) / 256, 256, 0, stream>>>(ftm_re, ftm_im, g_re, g_im);
  gemm_k<2><<<gblocks(64, 64, MB), GB, 0, stream>>>(
      Dfwd, nullptr, 64 * TE, TE, g_re, g_im, TE * 64, 64,
      flm_re, flm_im, 64, MB * 64, nullptr, 64, 64, TE, MB);
  gemm_k<2><<<gblocks(64, 64, MB), GB, 0, stream>>>(
      Dinv, nullptr, 64 * 64, 64, flm_re, flm_im, 64, MB * 64,
      ftm2_re, ftm2_im, 64, MB * 64, nullptr, 64, 64, 64, MB);
  gemm_k<4><<<gblocks(PO, 64, 64), GB, 0, stream>>>(
      Ci, Si, 0, MB, ftm2_re, ftm2_im, MB * 64, 64,
      hgrid, nullptr, PO * 64, 64, nullptr, PO, 64, MB, 64);
  // hfin = hgrid @ spat_conv2_w + (conv2_b + b_t)  (M=64*PO, K=64, N=64)
  gemm_k<0><<<gblocks(64 * PO, 64, 1), GB, 0, stream>>>(
      hgrid, nullptr, 0, 64, c2w, nullptr, 0, 64,
      hfin, nullptr, 0, 64, b2c, 64 * PO, 64, 64, 1);

  // ---- Fuse: gelu + layernorm -> compact output ----
  final_k<<<(64 * 127 + 7) / 8, 256, 0, stream>>>(xs, hfin, lns, lnb, out);
}